// NetPPRGATDouble_46033459478687
// MI455X (gfx1250) — compile-verified
//
#include <hip/hip_runtime.h>
#include <hip/hip_bf16.h>

typedef __attribute__((ext_vector_type(2))) float v2f;
typedef __attribute__((ext_vector_type(8))) float v8f;

#define HEADS 4
#define IN_CH 128
#define HID1 32
#define OUT_CH 40
#define NEG_SLOPE 0.2f

// ---------------------------------------------------------------------------
// WMMA f32 GEMM: C[M x Ncol] = A[M x K] * B[K x Ncol]
// One wave per 16-row block; NT = Ncol/16 column tiles accumulated in regs.
// A layout (32-bit A 16x4): lanes 0-15 -> K={k,k+1}, lanes 16-31 -> K={k+2,k+3}
// C/D layout: vgpr r -> row m0 + (lane>=16 ? 8 : 0) + r, col = tile*16 + lane%16
// ---------------------------------------------------------------------------
template <int NT>
__global__ void gemm_wmma_f32(const float* __restrict__ A,
                              const float* __restrict__ B,
                              float* __restrict__ C,
                              int M, int K, int Ncol) {
    int wave = (int)((blockIdx.x * blockDim.x + threadIdx.x) >> 5);
    int lane = threadIdx.x & 31;
    int m0 = wave * 16;
    if (m0 >= M) return;                 // wave-uniform: EXEC stays all-ones
    int half = lane >> 4;                // 0 or 1
    int l16  = lane & 15;

    v8f zero = {};
    v8f acc[NT];
#pragma unroll
    for (int t = 0; t < NT; ++t) acc[t] = zero;

    const float* arow = A + (size_t)(m0 + l16) * K + half * 2;
    for (int k = 0; k < K; k += 4) {
        v2f a;
        a.x = arow[k];
        a.y = arow[k + 1];
#pragma unroll
        for (int t = 0; t < NT; ++t) {
            int col = t * 16 + l16;
            v2f b;
            b.x = B[(size_t)(k + half * 2) * Ncol + col];
            b.y = B[(size_t)(k + half * 2 + 1) * Ncol + col];
            acc[t] = __builtin_amdgcn_wmma_f32_16x16x4_f32(
                false, a, false, b, (short)0, acc[t], false, false);
        }
    }
#pragma unroll
    for (int t = 0; t < NT; ++t) {
        int col = t * 16 + l16;
        float* crow = C + (size_t)(m0 + half * 8) * Ncol + col;
#pragma unroll
        for (int r = 0; r < 8; ++r) crow[(size_t)r * Ncol] = acc[t][r];
    }
}

// ---------------------------------------------------------------------------
// Per-(node, head) attention score dot products
// ---------------------------------------------------------------------------
__global__ void att_scores(const float* __restrict__ h,
                           const float* __restrict__ att_s,
                           const float* __restrict__ att_d,
                           float* __restrict__ asrc, float* __restrict__ adst,
                           int n, int C) {
    int i = blockIdx.x * blockDim.x + threadIdx.x;
    if (i >= n * HEADS) return;
    int node = i / HEADS, head = i % HEADS;
    const float* hp = h + (size_t)node * HEADS * C + head * C;
    const float* as = att_s + head * C;
    const float* ad = att_d + head * C;
    float s = 0.f, d = 0.f;
    for (int c = 0; c < C; ++c) {
        float v = hp[c];
        s += v * as[c];
        d += v * ad[c];
    }
    asrc[i] = s;
    adst[i] = d;
}

__global__ void fill_f32(float* p, float v, long n) {
    long i = (long)blockIdx.x * blockDim.x + threadIdx.x;
    long stride = (long)gridDim.x * blockDim.x;
    for (; i < n; i += stride) p[i] = v;
}

__device__ __forceinline__ float lrelu(float x) {
    return x > 0.f ? x : NEG_SLOPE * x;
}

__device__ __forceinline__ void atomicMaxF(float* addr, float val) {
    if (val >= 0.f)
        atomicMax((int*)addr, __float_as_int(val));
    else
        atomicMin((unsigned int*)addr, __float_as_uint(val));
}

// pass 1: segment max of leaky-relu logits per (dst, head)
__global__ void edge_max(const int* __restrict__ src, const int* __restrict__ dst,
                         const float* __restrict__ asrc, const float* __restrict__ adst,
                         float* __restrict__ m, int nE) {
    int i = blockIdx.x * blockDim.x + threadIdx.x;
    if (i >= nE * HEADS) return;
    int e = i >> 2, hh = i & 3;
    int s = src[e], d = dst[e];
    float ev = lrelu(asrc[s * HEADS + hh] + adst[d * HEADS + hh]);
    atomicMaxF(&m[d * HEADS + hh], ev);
}

// pass 2: softmax denominators
__global__ void edge_den(const int* __restrict__ src, const int* __restrict__ dst,
                         const float* __restrict__ asrc, const float* __restrict__ adst,
                         const float* __restrict__ m, float* __restrict__ den, int nE) {
    int i = blockIdx.x * blockDim.x + threadIdx.x;
    if (i >= nE * HEADS) return;
    int e = i >> 2, hh = i & 3;
    int s = src[e], d = dst[e];
    float ev = lrelu(asrc[s * HEADS + hh] + adst[d * HEADS + hh]);
    atomicAdd(&den[d * HEADS + hh], __expf(ev - m[d * HEADS + hh]));
}

// pass 3: wave per edge; lane l owns flat channels {l, l+32, ...} of H*C so the
// h[src] gather and the atomicAdd scatter to out[dst] are fully coalesced.
template <int C>
__global__ void edge_aggr(const int* __restrict__ src, const int* __restrict__ dst,
                          const float* __restrict__ asrc, const float* __restrict__ adst,
                          const float* __restrict__ m, const float* __restrict__ den,
                          const float* __restrict__ h, float* __restrict__ out,
                          float* __restrict__ alphaOut, int nE) {
    int wave = (int)((blockIdx.x * blockDim.x + threadIdx.x) >> 5);
    int lane = threadIdx.x & 31;
    if (wave >= nE) return;
    int s = src[wave], d = dst[wave];
    const int T = HEADS * C;
#pragma unroll
    for (int f = lane; f < T; f += 32) {
        int hh = f / C;
        float ev = lrelu(asrc[s * HEADS + hh] + adst[d * HEADS + hh]);
        float al = __expf(ev - m[d * HEADS + hh]) / den[d * HEADS + hh];
        atomicAdd(&out[(size_t)d * T + f], al * h[(size_t)s * T + f]);
    }
    if (lane < HEADS) {
        int hh = lane;
        float ev = lrelu(asrc[s * HEADS + hh] + adst[d * HEADS + hh]);
        float al = __expf(ev - m[d * HEADS + hh]) / den[d * HEADS + hh];
        alphaOut[(size_t)wave * HEADS + hh] = al;
    }
}

__global__ void add_bias_elu(float* __restrict__ o, const float* __restrict__ b,
                             int n, int T) {
    int i = blockIdx.x * blockDim.x + threadIdx.x;
    if (i >= n * T) return;
    float v = o[i] + b[i % T];
    o[i] = v > 0.f ? v : (__expf(v) - 1.f);
}

// head-mean + bias + log_softmax per node
__global__ void finalize(const float* __restrict__ o2, const float* __restrict__ b2,
                         float* __restrict__ logp, int n) {
    int node = blockIdx.x * blockDim.x + threadIdx.x;
    if (node >= n) return;
    const float* p = o2 + (size_t)node * HEADS * OUT_CH;
    float v[OUT_CH];
    float mx = -1e30f;
#pragma unroll
    for (int c = 0; c < OUT_CH; ++c) {
        float s = 0.f;
#pragma unroll
        for (int hh = 0; hh < HEADS; ++hh) s += p[hh * OUT_CH + c];
        float val = s * (1.f / HEADS) + b2[c];
        v[c] = val;
        mx = fmaxf(mx, val);
    }
    float se = 0.f;
#pragma unroll
    for (int c = 0; c < OUT_CH; ++c) se += __expf(v[c] - mx);
    float lse = mx + __logf(se);
#pragma unroll
    for (int c = 0; c < OUT_CH; ++c)
        logp[(size_t)node * OUT_CH + c] = v[c] - lse;
}

__global__ void copy_i32_as_f32(const int* __restrict__ src, float* __restrict__ dst,
                                long n) {
    long i = (long)blockIdx.x * blockDim.x + threadIdx.x;
    if (i < n) dst[i] = (float)src[i];
}

// ---------------------------------------------------------------------------
extern "C" void kernel_launch(void* const* d_in, const int* in_sizes, int n_in,
                              void* d_out, int out_size, void* d_ws, size_t ws_size,
                              hipStream_t stream) {
    const float* x        = (const float*)d_in[0];
    const int*   ei       = (const int*)d_in[1];   // [2, E] int32 (jax default)
    const float* W1       = (const float*)d_in[2];
    const float* att_src1 = (const float*)d_in[3];
    const float* att_dst1 = (const float*)d_in[4];
    const float* b1       = (const float*)d_in[5];
    const float* W2       = (const float*)d_in[6];
    const float* att_src2 = (const float*)d_in[7];
    const float* att_dst2 = (const float*)d_in[8];
    const float* b2       = (const float*)d_in[9];

    const int N = in_sizes[0] / IN_CH;   // 100000
    const int E = in_sizes[1] / 2;       // 1600000
    const int T1 = HEADS * HID1;         // 128
    const int T2 = HEADS * OUT_CH;       // 160
    const int* esrc = ei;
    const int* edst = ei + E;

    // workspace layout (floats)
    float* ws    = (float*)d_ws;
    float* bufH  = ws;                               // N*160 (h1 then h2)
    float* bufO1 = bufH  + (size_t)N * T2;           // N*128
    float* bufO2 = bufO1 + (size_t)N * T1;           // N*160
    float* asrc  = bufO2 + (size_t)N * T2;           // N*4
    float* adst  = asrc  + (size_t)N * HEADS;        // N*4
    float* mbuf  = adst  + (size_t)N * HEADS;        // N*4
    float* den   = mbuf  + (size_t)N * HEADS;        // N*4

    // output layout (floats, reference tuple flattened in return order)
    float* out_logp   = (float*)d_out;                     // N*40
    float* out_ei1    = out_logp + (size_t)N * OUT_CH;     // 2*E
    float* out_alpha1 = out_ei1  + (size_t)2 * E;          // E*4
    float* out_ei2    = out_alpha1 + (size_t)E * HEADS;    // 2*E
    float* out_alpha2 = out_ei2  + (size_t)2 * E;          // E*4

    const int BLK = 256;
    int gemm_blocks = ((N / 16) * 32 + BLK - 1) / BLK + 1;
    int nh_blocks   = (N * HEADS + BLK - 1) / BLK;
    int eh_blocks   = (E * HEADS + BLK - 1) / BLK;
    int ew_blocks   = (E * 32 + BLK - 1) / BLK;   // wave per edge

    // ---------------- Layer 1 ----------------
    gemm_wmma_f32<8><<<gemm_blocks, BLK, 0, stream>>>(x, W1, bufH, N, IN_CH, T1);
    att_scores<<<nh_blocks, BLK, 0, stream>>>(bufH, att_src1, att_dst1, asrc, adst, N, HID1);
    fill_f32<<<2048, BLK, 0, stream>>>(mbuf, -INFINITY, (long)N * HEADS);
    fill_f32<<<2048, BLK, 0, stream>>>(den, 0.f, (long)N * HEADS);
    fill_f32<<<4096, BLK, 0, stream>>>(bufO1, 0.f, (long)N * T1);
    edge_max<<<eh_blocks, BLK, 0, stream>>>(esrc, edst, asrc, adst, mbuf, E);
    edge_den<<<eh_blocks, BLK, 0, stream>>>(esrc, edst, asrc, adst, mbuf, den, E);
    edge_aggr<HID1><<<ew_blocks, BLK, 0, stream>>>(esrc, edst, asrc, adst, mbuf, den,
                                                   bufH, bufO1, out_alpha1, E);
    add_bias_elu<<<(N * T1 + BLK - 1) / BLK, BLK, 0, stream>>>(bufO1, b1, N, T1);

    // ---------------- Layer 2 ----------------
    gemm_wmma_f32<10><<<gemm_blocks, BLK, 0, stream>>>(bufO1, W2, bufH, N, T1, T2);
    att_scores<<<nh_blocks, BLK, 0, stream>>>(bufH, att_src2, att_dst2, asrc, adst, N, OUT_CH);
    fill_f32<<<2048, BLK, 0, stream>>>(mbuf, -INFINITY, (long)N * HEADS);
    fill_f32<<<2048, BLK, 0, stream>>>(den, 0.f, (long)N * HEADS);
    fill_f32<<<4096, BLK, 0, stream>>>(bufO2, 0.f, (long)N * T2);
    edge_max<<<eh_blocks, BLK, 0, stream>>>(esrc, edst, asrc, adst, mbuf, E);
    edge_den<<<eh_blocks, BLK, 0, stream>>>(esrc, edst, asrc, adst, mbuf, den, E);
    edge_aggr<OUT_CH><<<ew_blocks, BLK, 0, stream>>>(esrc, edst, asrc, adst, mbuf, den,
                                                     bufH, bufO2, out_alpha2, E);

    // ---------------- Output assembly ----------------
    finalize<<<(N + BLK - 1) / BLK, BLK, 0, stream>>>(bufO2, b2, out_logp, N);
    copy_i32_as_f32<<<(2 * E + BLK - 1) / BLK, BLK, 0, stream>>>(ei, out_ei1, (long)2 * E);
    copy_i32_as_f32<<<(2 * E + BLK - 1) / BLK, BLK, 0, stream>>>(ei, out_ei2, (long)2 * E);
}